// SpinOrbitNeuron_71571335020811
// MI455X (gfx1250) — compile-verified
//
#include <hip/hip_runtime.h>
#include <math.h>

typedef __attribute__((ext_vector_type(2))) float v2f;
typedef __attribute__((ext_vector_type(8))) float v8f;

#define PI_F     3.14159265358979323846f
#define TWO_PI_F 6.28318530717958647692f
#define ALPHA_F  0.02f
#define BETA_F   0.015f
#define E_EX_F   0.1f
#define ZEEMAN_F (5.79e-05f * 2.0f * 0.1f)
// SCALE = T_EVO / HBAR = 1e-11 / 6.582e-16
#define SCALE_F  ((float)(1e-11 / 6.582e-16))
// damp = -SCALE * DEPH, DEPH = 1e7 * 1e-11 = 1e-4
#define DAMP_F   ((float)(-(1e-11 / 6.582e-16) * 1e-4))

__device__ __forceinline__ float wrap_pi(float x) {
  // Python-style mod: result of fmod matched to divisor sign
  float r = fmodf(x + PI_F, TWO_PI_F);
  if (r < 0.0f) r += TWO_PI_F;
  return r - PI_F;
}

// Wave32 all-reduce sum via V_WMMA_F32_16X16X4_F32.
// A (16x4): lane L (0-15) supplies {K0=v, K1=0}, lane L+16 supplies {K2=v, K3=0}.
// B = ones(4x16)  =>  D[m][*] = v_m + v_{m+16} (row-sum replicated over columns).
// Each half-wave's 8 accumulator VGPRs hold rows 0-7 / 8-15; sum them and
// exchange across halves with one shfl_xor(16). Exact f32 math. EXEC must be
// all ones at the WMMA (callers are fully reconverged here).
__device__ __forceinline__ float wave_sum_wmma(float v) {
  v2f a;    a.x = v;    a.y = 0.0f;
  v2f ones; ones.x = 1.0f; ones.y = 1.0f;
  v8f c = {};
  c = __builtin_amdgcn_wmma_f32_16x16x4_f32(false, a, false, ones, (short)0, c,
                                            false, false);
  float s = ((c[0] + c[1]) + (c[2] + c[3])) + ((c[4] + c[5]) + (c[6] + c[7]));
  s += __shfl_xor(s, 16, 32);
  return s; // wave-uniform total over all 32 lanes
}

// Tiny kernel: fold exchange_fields (16x3) + gap into 3 field constants.
__global__ void son_const_kernel(const float* __restrict__ ex,
                                 const float* __restrict__ gap,
                                 float* __restrict__ cst) {
  if (threadIdx.x == 0) {
    float sx = 0.f, sy = 0.f, sz = 0.f;
    for (int i = 0; i < 16; ++i) {
      sx += ex[3 * i + 0];
      sy += ex[3 * i + 1];
      sz += ex[3 * i + 2];
    }
    cst[0] = E_EX_F * (sx * (1.0f / 16.0f));
    cst[1] = E_EX_F * (sy * (1.0f / 16.0f));
    cst[2] = E_EX_F * (sz * (1.0f / 16.0f)) + ZEEMAN_F + gap[0];
  }
}

__global__ __launch_bounds__(256) void son_main_kernel(
    const float* __restrict__ in,   // (B,16)
    const float* __restrict__ sre,  // (B,2)
    const float* __restrict__ sim,  // (B,2)
    const float* __restrict__ cst,  // [cJx, cJy, chz]
    float* __restrict__ out, int B,
    float* __restrict__ wsbp, float* __restrict__ wsdg,
    float* __restrict__ wsfirst, float* __restrict__ wslast) {
  __shared__ float sAng[256];
  __shared__ float sBp[8];
  __shared__ float sDg[8];

  const int  t = threadIdx.x;
  const long i = (long)blockIdx.x * 256 + t;
  const bool valid = i < (long)B;

  float bp_e = 0.0f, ang = 0.0f;

  if (valid) {
    v2f k = ((const v2f*)in)[i * 8];   // first 8 bytes of the 64B row
    v2f r = ((const v2f*)sre)[i];
    v2f m = ((const v2f*)sim)[i];

    // normalize spinor
    float n2  = r.x * r.x + m.x * m.x + r.y * r.y + m.y * m.y;
    float inv = rsqrtf(n2);
    float a_r = r.x * inv, a_i = m.x * inv;
    float b_r = r.y * inv, b_i = m.y * inv;

    // effective field
    float hx = fmaf(ALPHA_F, k.y, fmaf(BETA_F, k.x, cst[0]));
    float hy = fmaf(-ALPHA_F, k.x, fmaf(-BETA_F, k.y, cst[1]));
    float hz = cst[2];

    // closed-form exp of (-i*S*H - S*DEPH*I): SU(2) rotation * damping scalar
    float E  = sqrtf(hx * hx + hy * hy + hz * hz);
    float th = SCALE_F * E;
    float sn, cs;
    sincosf(th, &sn, &cs);
    float sinc = (th < 1e-12f) ? 1.0f : sn / th;
    float f  = sinc * SCALE_F;
    float ed = expf(DAMP_F);           // constant-folded
    float fx = f * hx, fy = f * hy, fz = f * hz;

    // p = U * psi0 ; U = ed * [[cs - i fz, -fy - i fx], [fy - i fx, cs + i fz]]
    float p0r = ed * ((cs * a_r + fz * a_i) + (-fy * b_r + fx * b_i));
    float p0i = ed * ((cs * a_i - fz * a_r) + (-fy * b_i - fx * b_r));
    float p1r = ed * ((fy * a_r + fx * a_i) + (cs * b_r - fz * b_i));
    float p1i = ed * ((fy * a_i - fx * a_r) + (cs * b_i + fz * b_r));

    float ph0 = atan2f(p0i, p0r);
    float ph1 = atan2f(p1i, p1r);
    bp_e = ph1 - ph0;

    // spin expectation from cross = p0 * conj(p1)
    float cr  = p0r * p1r + p0i * p1i;
    float ci  = p0i * p1r - p0r * p1i;
    float sxv = 2.0f * cr;
    float syv = 2.0f * ci;
    float szv = (p0r * p0r + p0i * p0i) - (p1r * p1r + p1i * p1i);
    ang = atan2f(syv, sxv);

    // --- outputs ---
    out[3 * i + 0] = sxv;
    out[3 * i + 1] = syv;
    out[3 * i + 2] = szv;
    float4 psi = make_float4(p0r, p0i, p1r, p1i);      // complex64 (B,2) interleaved
    ((float4*)(out + 3L * B))[i] = psi;
    v2f mom; mom.x = k.x; mom.y = k.y;
    ((v2f*)(out + 7L * B))[i] = mom;                   // momentum
  }

  // winding-angle neighbor differences within the block
  sAng[t] = ang;
  __syncthreads();
  float dg = 0.0f;
  if (valid && t < 255 && (i + 1) < (long)B) dg = wrap_pi(sAng[t + 1] - ang);

  // wave-level exact f32 all-reduce via WMMA, then cross-wave via LDS
  float wbp = wave_sum_wmma(bp_e);
  float wdg = wave_sum_wmma(dg);
  int wid = t >> 5, lid = t & 31;
  if (lid == 0) { sBp[wid] = wbp; sDg[wid] = wdg; }
  __syncthreads();
  if (t == 0) {
    float b = 0.f, d = 0.f;
    for (int w = 0; w < 8; ++w) { b += sBp[w]; d += sDg[w]; }
    wsbp[blockIdx.x] = b;
    wsdg[blockIdx.x] = d;
    wsfirst[blockIdx.x] = sAng[0];
  }
  if (t == 255) wslast[blockIdx.x] = ang;   // only consumed for non-final full blocks
}

__global__ __launch_bounds__(256) void son_finish_kernel(
    const float* __restrict__ wsbp, const float* __restrict__ wsdg,
    const float* __restrict__ wsfirst, const float* __restrict__ wslast,
    int nblk, int B, float* __restrict__ out) {
  __shared__ float sB[8];
  __shared__ float sD[8];
  const int t = threadIdx.x;
  float b = 0.f, d = 0.f;
  for (int idx = t; idx < nblk; idx += 256) { b += wsbp[idx]; d += wsdg[idx]; }
  for (int idx = t; idx < nblk - 1; idx += 256)
    d += wrap_pi(wsfirst[idx + 1] - wslast[idx]);   // stitch block boundaries

  float wb = wave_sum_wmma(b);
  float wd = wave_sum_wmma(d);
  if ((t & 31) == 0) { sB[t >> 5] = wb; sD[t >> 5] = wd; }
  __syncthreads();
  if (t == 0) {
    float tb = 0.f, td = 0.f;
    for (int w = 0; w < 8; ++w) { tb += sB[w]; td += sD[w]; }
    float berry = wrap_pi(tb / (float)B);
    int charge = __float2int_rn(td * (1.0f / TWO_PI_F));   // round-half-even
    out[9L * B] = berry;
    ((int*)out)[9L * B + 1] = charge;                      // int32 bit pattern
  }
}

extern "C" void kernel_launch(void* const* d_in, const int* in_sizes, int n_in,
                              void* d_out, int out_size, void* d_ws, size_t ws_size,
                              hipStream_t stream) {
  (void)n_in; (void)out_size; (void)ws_size;
  const float* in0 = (const float*)d_in[0];  // input_data  (B,16)
  const float* sre = (const float*)d_in[1];  // spinor_re   (B,2)
  const float* sim = (const float*)d_in[2];  // spinor_im   (B,2)
  const float* ex  = (const float*)d_in[3];  // exchange_fields (16,3)
  const float* gap = (const float*)d_in[4];  // topological_gap (1,)

  const int B    = in_sizes[0] / 16;
  const int nblk = (B + 255) / 256;

  float* ws      = (float*)d_ws;
  float* wsbp    = ws;                 // per-block berry-phase partial sums
  float* wsdg    = ws + nblk;          // per-block interior winding sums
  float* wsfirst = ws + 2 * nblk;      // first ang of each block
  float* wslast  = ws + 3 * nblk;      // last ang of each block
  float* cst     = ws + 4 * nblk;      // [cJx, cJy, chz]

  son_const_kernel<<<1, 32, 0, stream>>>(ex, gap, cst);
  son_main_kernel<<<nblk, 256, 0, stream>>>(in0, sre, sim, cst, (float*)d_out, B,
                                            wsbp, wsdg, wsfirst, wslast);
  son_finish_kernel<<<1, 256, 0, stream>>>(wsbp, wsdg, wsfirst, wslast, nblk, B,
                                           (float*)d_out);
}